// SensorGCNEncoder_64338610095072
// MI455X (gfx1250) — compile-verified
//
#include <hip/hip_runtime.h>
#include <hip/hip_bf16.h>

// Problem constants (from reference)
#define T_SEQ   8192
#define N_BATCH 32
#define IN_D    6
#define D1      12
#define D2      12
#define D3      24
#define LATENT  256
#define EPSF    1e-5f

// Tiling
#define TILE    256
#define HALO    3
#define NODES   (TILE + 2 * HALO)   // 262
#define NTHREADS 256

// LDS strides (chosen for bank-conflict freedom on 64 banks)
#define HSTR    13                  // 12-feature rows
#define H3STR   25                  // 24-feature rows
#define WO_STR  264                 // Wo rows: 2*264 % 64 == 16 -> disjoint bank windows

typedef __attribute__((ext_vector_type(2))) float v2f;
typedef __attribute__((ext_vector_type(8))) float v8f;

__device__ __forceinline__ float dinv_of(int t) {
    if (t < 0 || t >= T_SEQ) return 0.0f;
    return (t == 0 || t == T_SEQ - 1) ? 0.70710678118654752f   // 1/sqrt(2)
                                      : 0.57735026918962576f;  // 1/sqrt(3)
}

// hw = h @ W for nodes i in [lo, hi)
template <int DI, int DO>
__device__ __forceinline__ void node_matmul(const float* __restrict__ in, float* __restrict__ outb,
                                            int strIn, int strOut, int lo, int hi,
                                            const float* __restrict__ W, int tid) {
    for (int i = lo + tid; i < hi; i += NTHREADS) {
        float xv[DI];
#pragma unroll
        for (int c = 0; c < DI; ++c) xv[c] = in[i * strIn + c];
#pragma unroll
        for (int f = 0; f < DO; ++f) {
            float s = 0.0f;
#pragma unroll
            for (int c = 0; c < DI; ++c) s += xv[c] * W[c * DO + f];
            outb[i * strOut + f] = s;
        }
    }
}

// 3-tap normalized chain conv + bias + LayerNorm + ReLU, nodes i in [lo, hi)
template <int D>
__device__ __forceinline__ void conv_ln_relu(const float* __restrict__ in, float* __restrict__ outb,
                                             int strIn, int strOut, int lo, int hi, int t0,
                                             const float* __restrict__ b, const float* __restrict__ g,
                                             const float* __restrict__ be, int tid) {
    for (int i = lo + tid; i < hi; i += NTHREADS) {
        int t = t0 - HALO + i;
        float v[D];
        if (t >= 0 && t < T_SEQ) {
            float di = dinv_of(t);
            float dm = dinv_of(t - 1);
            float dp = dinv_of(t + 1);
            float mu = 0.0f;
#pragma unroll
            for (int f = 0; f < D; ++f) {
                float a = di * (di * in[i * strIn + f] +
                                dm * in[(i - 1) * strIn + f] +
                                dp * in[(i + 1) * strIn + f]) + b[f];
                v[f] = a;
                mu += a;
            }
            mu *= (1.0f / D);
            float var = 0.0f;
#pragma unroll
            for (int f = 0; f < D; ++f) { float d = v[f] - mu; var += d * d; }
            var *= (1.0f / D);
            float rs = rsqrtf(var + EPSF);
#pragma unroll
            for (int f = 0; f < D; ++f) {
                float y = (v[f] - mu) * rs * g[f] + be[f];
                v[f] = y > 0.0f ? y : 0.0f;
            }
        } else {
#pragma unroll
            for (int f = 0; f < D; ++f) v[f] = 0.0f;
        }
#pragma unroll
        for (int f = 0; f < D; ++f) outb[i * strOut + f] = v[f];
    }
}

__launch_bounds__(NTHREADS)
__global__ void gcn_chain_encoder_fused(
    const float* __restrict__ x,
    const float* __restrict__ W1, const float* __restrict__ b1,
    const float* __restrict__ W2, const float* __restrict__ b2,
    const float* __restrict__ W3, const float* __restrict__ b3,
    const float* __restrict__ g1, const float* __restrict__ be1,
    const float* __restrict__ g2, const float* __restrict__ be2,
    const float* __restrict__ g3, const float* __restrict__ be3,
    const float* __restrict__ Wo, const float* __restrict__ bo,
    float* __restrict__ out) {

    __shared__ float sW1[IN_D * D1];
    __shared__ float sW2[D1 * D2];
    __shared__ float sW3[D2 * D3];
    __shared__ float sb1[D1], sg1[D1], sbe1[D1];
    __shared__ float sb2[D2], sg2[D2], sbe2[D2];
    __shared__ float sb3[D3], sg3[D3], sbe3[D3];
    __shared__ float sbo[LATENT];
    __shared__ float sWo[D3 * WO_STR];        // 24 x 264
    __shared__ float hA[NODES * HSTR];        // 262 x 13
    __shared__ float hB[NODES * HSTR];        // 262 x 13
    __shared__ float h3a[NODES * H3STR];      // 262 x 25
    __shared__ float h3b[TILE * H3STR];       // 256 x 25

    const int tid = threadIdx.x;
    const int bid = blockIdx.x;
    const int seq = bid / (T_SEQ / TILE);
    const int t0  = (bid % (T_SEQ / TILE)) * TILE;
    const int node_base = seq * T_SEQ;        // first node of this sequence

    // ---- stage 0: stage params into LDS ----
    for (int i = tid; i < IN_D * D1; i += NTHREADS) sW1[i] = W1[i];
    for (int i = tid; i < D1 * D2; i += NTHREADS)   sW2[i] = W2[i];
    for (int i = tid; i < D2 * D3; i += NTHREADS)   sW3[i] = W3[i];
    if (tid < D1) { sb1[tid] = b1[tid]; sg1[tid] = g1[tid]; sbe1[tid] = be1[tid]; }
    if (tid < D2) { sb2[tid] = b2[tid]; sg2[tid] = g2[tid]; sbe2[tid] = be2[tid]; }
    if (tid < D3) { sb3[tid] = b3[tid]; sg3[tid] = g3[tid]; sbe3[tid] = be3[tid]; }
    sbo[tid] = bo[tid];
    for (int i = tid; i < D3 * LATENT; i += NTHREADS) {
        int k = i >> 8, n = i & (LATENT - 1);
        sWo[k * WO_STR + n] = Wo[i];
    }

    // ---- stage 1: hw1 = x @ W1 over halo range [0, NODES) ----
    for (int i = tid; i < NODES; i += NTHREADS) {
        int t = t0 - HALO + i;
        float hv[D1];
        if (t >= 0 && t < T_SEQ) {
            const float* xp = x + (size_t)(node_base + t) * IN_D;
            float xv[IN_D];
#pragma unroll
            for (int c = 0; c < IN_D; ++c) xv[c] = xp[c];
#pragma unroll
            for (int f = 0; f < D1; ++f) {
                float s = 0.0f;
#pragma unroll
                for (int c = 0; c < IN_D; ++c) s += xv[c] * sW1[c * D1 + f];
                hv[f] = s;
            }
        } else {
#pragma unroll
            for (int f = 0; f < D1; ++f) hv[f] = 0.0f;
        }
#pragma unroll
        for (int f = 0; f < D1; ++f) hA[i * HSTR + f] = hv[f];
    }
    __syncthreads();

    // ---- layer 1 conv+LN+ReLU: hA -> hB, valid [1, 261) ----
    conv_ln_relu<D1>(hA, hB, HSTR, HSTR, 1, NODES - 1, t0, sb1, sg1, sbe1, tid);
    __syncthreads();

    // ---- layer 2: hw2 = h1 @ W2 (hB -> hA), then conv (hA -> hB) ----
    node_matmul<D1, D2>(hB, hA, HSTR, HSTR, 1, NODES - 1, sW2, tid);
    __syncthreads();
    conv_ln_relu<D2>(hA, hB, HSTR, HSTR, 2, NODES - 2, t0, sb2, sg2, sbe2, tid);
    __syncthreads();

    // ---- layer 3: hw3 = h2 @ W3 (hB -> h3a), then conv (h3a -> h3b central) ----
    node_matmul<D2, D3>(hB, h3a, HSTR, H3STR, 2, NODES - 2, sW3, tid);
    __syncthreads();
    conv_ln_relu<D3>(h3a, h3b - HALO * H3STR, H3STR, H3STR, HALO, NODES - HALO, t0,
                     sb3, sg3, sbe3, tid);
    __syncthreads();

    // ---- stage 4: out = h3 @ Wo + bo via V_WMMA_F32_16X16X4_F32 ----
    const int lane = tid & 31;
    const int wav  = tid >> 5;
    const int l15  = lane & 15;
    const int half = lane >> 4;       // 0: lanes 0-15, 1: lanes 16-31

    for (int nt = wav; nt < LATENT / 16; nt += NTHREADS / 32) {
        const int n0 = nt * 16;
        // B fragments for all 6 K-steps (K = 24, steps of 4), reused across 16 M-tiles.
        // B 4x16 f32: VGPR0 = rows {K+0 | K+2}, VGPR1 = rows {K+1 | K+3} (lower|upper lanes)
        v2f bf[6];
#pragma unroll
        for (int kk = 0; kk < 6; ++kk) {
            int kb = kk * 4 + 2 * half;
            bf[kk].x = sWo[kb * WO_STR + n0 + l15];
            bf[kk].y = sWo[(kb + 1) * WO_STR + n0 + l15];
        }
        const float bov = sbo[n0 + l15];

        for (int mt = 0; mt < TILE / 16; ++mt) {
            v8f acc = {0.f, 0.f, 0.f, 0.f, 0.f, 0.f, 0.f, 0.f};
            const int mrow = mt * 16 + l15;
#pragma unroll
            for (int kk = 0; kk < 6; ++kk) {
                int kb = kk * 4 + 2 * half;
                v2f af;
                // A 16x4 f32: lanes 0-15 hold {K=0,K=1}, lanes 16-31 hold {K=2,K=3}
                af.x = h3b[mrow * H3STR + kb];
                af.y = h3b[mrow * H3STR + kb + 1];
                acc = __builtin_amdgcn_wmma_f32_16x16x4_f32(
                    /*neg_a=*/false, af, /*neg_b=*/false, bf[kk],
                    /*c_mod=*/(short)0, acc, /*reuse_a=*/false, /*reuse_b=*/false);
            }
            // C/D layout: VGPR r = row (r + 8*half), col = lane&15
            float* op = out + (size_t)(node_base + t0 + mt * 16) * LATENT + n0 + l15;
#pragma unroll
            for (int r = 0; r < 8; ++r) {
                op[(size_t)(r + 8 * half) * LATENT] = acc[r] + bov;
            }
        }
    }
}

extern "C" void kernel_launch(void* const* d_in, const int* in_sizes, int n_in,
                              void* d_out, int out_size, void* d_ws, size_t ws_size,
                              hipStream_t stream) {
    // setup_inputs order:
    // 0:x 1:ei 2:W1 3:b1 4:W2 5:b2 6:W3 7:b3 8:g1 9:be1 10:g2 11:be2 12:g3 13:be3 14:Wo 15:bo
    const float* x   = (const float*)d_in[0];
    const float* W1  = (const float*)d_in[2];
    const float* b1  = (const float*)d_in[3];
    const float* W2  = (const float*)d_in[4];
    const float* b2  = (const float*)d_in[5];
    const float* W3  = (const float*)d_in[6];
    const float* b3  = (const float*)d_in[7];
    const float* g1  = (const float*)d_in[8];
    const float* be1 = (const float*)d_in[9];
    const float* g2  = (const float*)d_in[10];
    const float* be2 = (const float*)d_in[11];
    const float* g3  = (const float*)d_in[12];
    const float* be3 = (const float*)d_in[13];
    const float* Wo  = (const float*)d_in[14];
    const float* bo  = (const float*)d_in[15];
    float* out = (float*)d_out;

    dim3 grid(N_BATCH * (T_SEQ / TILE));   // 32 * 32 = 1024 workgroups
    dim3 block(NTHREADS);
    gcn_chain_encoder_fused<<<grid, block, 0, stream>>>(
        x, W1, b1, W2, b2, W3, b3, g1, be1, g2, be2, g3, be3, Wo, bo, out);
}